// CenterLoss_15917148799608
// MI455X (gfx1250) — compile-verified
//
#include <hip/hip_runtime.h>

#define BATCH 65536
#define FEA 512
#define NUM_CLASSES 1000
#define ROWS_PER_BLOCK 64
#define NBLOCKS (BATCH / ROWS_PER_BLOCK)        // 1024
#define THREADS 256
#define STAGE_ROWS 8
#define STAGES (ROWS_PER_BLOCK / STAGE_ROWS)    // 8
#define STAGE_FLOATS (STAGE_ROWS * FEA)         // 4096 floats = 16 KB
#define NPS 4                                   // float4 async loads per thread per stage

#if defined(__has_builtin)
#if __has_builtin(__builtin_amdgcn_global_load_async_to_lds_b128)
#define HAVE_ASYNC_LDS 1
#endif
#endif
#ifndef HAVE_ASYNC_LDS
#define HAVE_ASYNC_LDS 0
#endif

typedef int v4i __attribute__((ext_vector_type(4)));
typedef __attribute__((address_space(1))) v4i* gptr_v4i;
typedef __attribute__((address_space(3))) v4i* lptr_v4i;

// One 128-bit async global->LDS load per lane (GLOBAL_LOAD_ASYNC_TO_LDS_B128, ASYNCcnt).
__device__ __forceinline__ void async_load_b128(const float* gsrc, float* ldst) {
#if HAVE_ASYNC_LDS
  __builtin_amdgcn_global_load_async_to_lds_b128(
      (gptr_v4i)gsrc, (lptr_v4i)ldst, /*offset=*/0, /*cpol=*/0);
#else
  *(float4*)ldst = *(const float4*)gsrc;
#endif
}

template <int N>
__device__ __forceinline__ void wait_async() {
#if HAVE_ASYNC_LDS
#if __has_builtin(__builtin_amdgcn_s_wait_asynccnt)
  __builtin_amdgcn_s_wait_asynccnt(N);
#else
  asm volatile("s_wait_asynccnt %0" ::"i"(N) : "memory");
#endif
#endif
}

__global__ __launch_bounds__(THREADS) void center_loss_partial(
    const float* __restrict__ x, const int* __restrict__ labels,
    const float* __restrict__ centers, float* __restrict__ part) {
  __shared__ float buf[3][STAGE_FLOATS];   // 3 x 16 KB rotating buffers for x tiles
  __shared__ int labs[ROWS_PER_BLOCK];
  __shared__ float red[THREADS];

  const int t = threadIdx.x;
  const int blockRow = blockIdx.x * ROWS_PER_BLOCK;
  const float* xblk = x + (size_t)blockRow * FEA;
  const int rowInPair = t >> 7;            // 0..1 within a 2-row chunk
  const int col = (t & 127) << 2;          // float4 column within the row

  // Preload this block's 64 labels into LDS (visible after the first stage barrier).
  if (t < ROWS_PER_BLOCK) labs[t] = labels[blockRow + t];

  // Prime the pipeline: stage 0.
#pragma unroll
  for (int k = 0; k < NPS; ++k) {
    const int idx = t * 4 + k * (THREADS * 4);
    async_load_b128(xblk + idx, &buf[0][idx]);
  }

  float acc = 0.0f;
#pragma unroll
  for (int s = 0; s < STAGES; ++s) {
    if (s + 1 < STAGES) {
      // Issue stage s+1 into buf[(s+1)%3]; last read by compute(s-2), fenced by
      // the iteration s-1 barrier -> safe with a single barrier per stage.
      const float* g = xblk + (size_t)(s + 1) * STAGE_FLOATS;
      float* l = buf[(s + 1) % 3];
#pragma unroll
      for (int k = 0; k < NPS; ++k) {
        const int idx = t * 4 + k * (THREADS * 4);
        async_load_b128(g + idx, l + idx);
      }
      wait_async<NPS>();   // async ops complete in order: stage s is fully in LDS
    } else {
      wait_async<0>();
    }
    __syncthreads();       // all waves' stage-s data visible (and labels on s==0)

    const float* lbuf = buf[s % 3];
#pragma unroll
    for (int k = 0; k < NPS; ++k) {
      const int idx = t * 4 + k * (THREADS * 4);       // flat float index in stage
      const int rowInStage = k * 2 + rowInPair;        // idx >> 9
      const int lab = labs[s * STAGE_ROWS + rowInStage];
      const float4 cv = *(const float4*)(centers + (size_t)lab * FEA + col);
      const float4 xv = *(const float4*)(&lbuf[idx]);
      const float dx = xv.x - cv.x;
      const float dy = xv.y - cv.y;
      const float dz = xv.z - cv.z;
      const float dw = xv.w - cv.w;
      acc += dx * dx + dy * dy + dz * dz + dw * dw;
    }
  }

  // Block reduction -> one partial per block (deterministic order).
  red[t] = acc;
  __syncthreads();
#pragma unroll
  for (int off = THREADS / 2; off > 0; off >>= 1) {
    if (t < off) red[t] += red[t + off];
    __syncthreads();
  }
  if (t == 0) part[blockIdx.x] = red[0];
}

__global__ __launch_bounds__(512) void reduce_partials(const float* __restrict__ part,
                                                       float* __restrict__ out) {
  __shared__ float red[512];
  const int t = threadIdx.x;
  red[t] = part[t] + part[t + 512];
  __syncthreads();
#pragma unroll
  for (int off = 256; off > 0; off >>= 1) {
    if (t < off) red[t] += red[t + off];
    __syncthreads();
  }
  if (t == 0) out[0] = red[0] * (0.5f / (float)BATCH);
}

extern "C" void kernel_launch(void* const* d_in, const int* in_sizes, int n_in,
                              void* d_out, int out_size, void* d_ws, size_t ws_size,
                              hipStream_t stream) {
  const float* x = (const float*)d_in[0];
  const int* labels = (const int*)d_in[1];
  const float* centers = (const float*)d_in[2];
  float* out = (float*)d_out;
  float* part = (float*)d_ws;  // NBLOCKS floats of scratch

  center_loss_partial<<<NBLOCKS, THREADS, 0, stream>>>(x, labels, centers, part);
  reduce_partials<<<1, 512, 0, stream>>>(part, out);
}